// Loss_32753420599585
// MI455X (gfx1250) — compile-verified
//
#include <hip/hip_runtime.h>
#include <stdint.h>

#define GRID7 7
#define CELLS 49            // 7*7
#define NCH 30
#define ITEM 1470           // 30*49 floats per batch item
#define BPB 2               // batch items per block (slab 11760 B, 16B aligned)
#define TPB 128             // 4 waves; small LDS footprint -> >=2 blocks/CU resident
#define TPB2 256            // finalize kernel

// ---- CDNA5 async global->LDS path (codegen-confirmed in round 2) --------
//   void __builtin_amdgcn_global_load_async_to_lds_b128(v4i AS1*, v4i AS3*, imm, imm)
typedef int v4i __attribute__((ext_vector_type(4)));
typedef __attribute__((address_space(1))) v4i* as1_v4i_ptr;
typedef __attribute__((address_space(3))) v4i* as3_v4i_ptr;

#if __has_builtin(__builtin_amdgcn_global_load_async_to_lds_b128) && \
    __has_builtin(__builtin_amdgcn_s_wait_asynccnt)
#define HAVE_ASYNC 1
#else
#define HAVE_ASYNC 0
#endif

__device__ __forceinline__ void copy16_to_lds(float* lds_dst, const float* g_src) {
#if HAVE_ASYNC
    // integer round-trips: generic global ptr is bit-identical to AS1; the low
    // 32 bits of a generic LDS ptr are the AS3 byte offset.
    __builtin_amdgcn_global_load_async_to_lds_b128(
        (as1_v4i_ptr)(uintptr_t)(const void*)g_src,
        (as3_v4i_ptr)(uint32_t)(uintptr_t)(void*)lds_dst,
        0, 0);
#else
    *(float4*)lds_dst = *(const float4*)g_src;
#endif
}

__device__ __forceinline__ void async_fence_all() {
#if HAVE_ASYNC
    __builtin_amdgcn_s_wait_asynccnt(0);
#endif
}

// ---- IOU faithful to reference (incl. disjoint->0 and inter<=0 -> 0) ----
__device__ __forceinline__ float iou_f(float x1a, float y1a, float x2a, float y2a,
                                       float x1b, float y1b, float x2b, float y2b) {
    bool sep = (x1a > x2b) | (y1a > y2b) | (x1b > x2a) | (y1b > y2a);
    float iw = fminf(x2a, x2b) - fmaxf(x1a, x1b);
    float ih = fminf(y2a, y2b) - fmaxf(y1a, y1b);
    float inter = iw * ih;
    float area1 = (x2a - x1a) * (y2a - y1a);
    float area2 = (x2b - x1b) * (y2b - y1b);
    float r = (inter > 0.0f) ? inter / (area1 + area2 - inter) : 0.0f;
    return sep ? 0.0f : r;
}

__global__ __launch_bounds__(TPB)
void yolo_loss_stage1(const float* __restrict__ pred,
                      const float* __restrict__ labels,
                      float* __restrict__ partial, int nbatch) {
    __shared__ float sP[BPB * ITEM];
    __shared__ float sL[BPB * ITEM];
    __shared__ float sRed[TPB / 32];

    const int tid = threadIdx.x;
    const int item0 = blockIdx.x * BPB;
    const int nitems = min(BPB, nbatch - item0);
    const int nfloats = nitems * ITEM;
    const float* gp = pred   + (size_t)item0 * ITEM;
    const float* gl = labels + (size_t)item0 * ITEM;

    // Bulk DMA both slabs into LDS: 16 B per lane per async op, fully coalesced,
    // tracked on ASYNCcnt (no VGPR staging).
    const int full = nfloats >> 2;            // 735 for full blocks (exact, no tail)
    for (int idx = tid; idx < full; idx += TPB) {
        copy16_to_lds(&sP[idx * 4], gp + idx * 4);
        copy16_to_lds(&sL[idx * 4], gl + idx * 4);
    }
    if (tid == 0) {                           // generic tail (never taken for B%2==0)
        for (int r = full * 4; r < nfloats; ++r) { sP[r] = gp[r]; sL[r] = gl[r]; }
    }
    async_fence_all();                        // s_wait_asynccnt 0
    __syncthreads();

    float acc = 0.0f;
    const int ncells = nitems * CELLS;
    for (int c = tid; c < ncells; c += TPB) {
        const int item = c / CELLS;
        const int ij = c - item * CELLS;
        const int i = ij / GRID7;             // row -> m
        const int j = ij - i * GRID7;         // col -> n
        const float* P = &sP[item * ITEM + ij];
        const float* L = &sL[item * ITEM + ij];
        const float m = (float)i;
        const float n = (float)j;

        float p0 = P[0*CELLS], p1 = P[1*CELLS], p2 = P[2*CELLS], p3 = P[3*CELLS], p4 = P[4*CELLS];
        float p5 = P[5*CELLS], p6 = P[6*CELLS], p7 = P[7*CELLS], p8 = P[8*CELLS], p9 = P[9*CELLS];
        float l0 = L[0*CELLS], l1 = L[1*CELLS], l2 = L[2*CELLS], l3 = L[3*CELLS], l4 = L[4*CELLS];
        float l5 = L[5*CELLS], l6 = L[6*CELLS], l7 = L[7*CELLS], l8 = L[8*CELLS];

        // predicted boxes
        float px = (p0 + m) / 7.0f, py = (p1 + n) / 7.0f;
        float b1x1 = px - p2 * 0.5f, b1y1 = py - p3 * 0.5f;
        float b1x2 = px + p2 * 0.5f, b1y2 = py + p3 * 0.5f;
        float qx = (p5 + m) / 7.0f, qy = (p6 + n) / 7.0f;
        float b2x1 = qx - p7 * 0.5f, b2y1 = qy - p8 * 0.5f;
        float b2x2 = qx + p7 * 0.5f, b2y2 = qy + p8 * 0.5f;

        // ground-truth box, faithfully buggy: y-center uses +m, and /7 for both
        float gcx = (l0 + m) / 7.0f;
        float gcy = (l1 + m) / 7.0f;
        float gx1 = gcx - l2 * 0.5f, gx2 = gcx + l2 * 0.5f;
        float gy1 = gcy - l3 * 0.5f, gy2 = gcy + l3 * 0.5f;

        float iou1 = iou_f(b1x1, b1y1, b1x2, b1y2, gx1, gy1, gx2, gy2);
        float iou2 = iou_f(b2x1, b2y1, b2x2, b2y2, gx1, gy1, gx2, gy2);

        float d0 = p0 - l0, d1 = p1 - l1;
        float s2 = sqrtf(p2) - sqrtf(l2), s3 = sqrtf(p3) - sqrtf(l3);
        float loc1 = 5.0f * (d0 * d0 + d1 * d1 + s2 * s2 + s3 * s3);
        float e5 = p5 - l5, e6 = p6 - l6;
        float s7 = sqrtf(p7) - sqrtf(l7), s8 = sqrtf(p8) - sqrtf(l8);
        float loc2 = 5.0f * (e5 * e5 + e6 * e6 + s7 * s7 + s8 * s8);

        float a4 = p4 - iou1, a9 = p9 - iou2;
        float q4 = a4 * a4, q9 = a9 * a9;
        float obj1 = q4 + 0.5f * q9;
        float obj2 = q9 + 0.5f * q4;

        float cls = 0.0f;
#pragma unroll
        for (int cc = 10; cc < NCH; ++cc) {
            float d = P[cc * CELLS] - L[cc * CELLS];
            cls += d * d;
        }

        float cell_obj   = ((iou1 >= iou2) ? (loc1 + obj1) : (loc2 + obj2)) + cls;
        float cell_noobj = 0.5f * (p4 * p4 + p9 * p9);
        acc += (l4 == 1.0f) ? cell_obj : cell_noobj;
    }

    // wave32 tree reduce, then cross-wave via LDS (fixed order -> deterministic)
    for (int off = 16; off > 0; off >>= 1)
        acc += __shfl_down(acc, off, 32);
    if ((tid & 31) == 0) sRed[tid >> 5] = acc;
    __syncthreads();
    if (tid == 0) {
        float s = 0.0f;
#pragma unroll
        for (int w = 0; w < TPB / 32; ++w) s += sRed[w];
        partial[blockIdx.x] = s;
    }
}

__global__ __launch_bounds__(TPB2)
void yolo_loss_stage2(const float* __restrict__ partial, int nparts,
                      float* __restrict__ out, float inv_b) {
    __shared__ float sRed[TPB2 / 32];
    const int tid = threadIdx.x;
    float s = 0.0f;
    for (int i = tid; i < nparts; i += TPB2) s += partial[i];  // fixed order
    for (int off = 16; off > 0; off >>= 1)
        s += __shfl_down(s, off, 32);
    if ((tid & 31) == 0) sRed[tid >> 5] = s;
    __syncthreads();
    if (tid == 0) {
        float t = 0.0f;
#pragma unroll
        for (int w = 0; w < TPB2 / 32; ++w) t += sRed[w];
        out[0] = t * inv_b;
    }
}

extern "C" void kernel_launch(void* const* d_in, const int* in_sizes, int n_in,
                              void* d_out, int out_size, void* d_ws, size_t ws_size,
                              hipStream_t stream) {
    const float* pred   = (const float*)d_in[0];
    const float* labels = (const float*)d_in[1];
    float* out = (float*)d_out;
    float* ws  = (float*)d_ws;

    const int nbatch  = in_sizes[0] / ITEM;            // 16384
    const int nblocks = (nbatch + BPB - 1) / BPB;      // 8192

    yolo_loss_stage1<<<nblocks, TPB, 0, stream>>>(pred, labels, ws, nbatch);
    yolo_loss_stage2<<<1, TPB2, 0, stream>>>(ws, nblocks, out, 1.0f / (float)nbatch);
}